// TGCN_67327907332470
// MI455X (gfx1250) — compile-verified
//
#include <hip/hip_runtime.h>
#include <hip/hip_bf16.h>

typedef __attribute__((ext_vector_type(2))) float v2f;
typedef __attribute__((ext_vector_type(8))) float v8f;

#define N_NODES 8192
#define N_EDGES 262144
#define D_IN 128
#define D_OUT 128
#define H_DIM 128
#define G3H 384   // 3*H

// ---------------------------------------------------------------------------
// GEMM 1: xw[N,128] = x[N,128] @ W[128,128]   (fp32 WMMA 16x16x4)
// One wave per 16x16 output tile. 8 waves / block.
// ---------------------------------------------------------------------------
__global__ void __launch_bounds__(256)
gemm_xw_kernel(const float* __restrict__ x, const float* __restrict__ W,
               float* __restrict__ xw) {
    const int lane  = threadIdx.x & 31;
    const int wave  = threadIdx.x >> 5;
    const int tile  = blockIdx.x * 8 + wave;
    const int nTiles = D_OUT / 16;              // 8
    const int mTile = tile / nTiles;
    const int nTile = tile % nTiles;
    const int mBase = mTile * 16;
    const int nBase = nTile * 16;

    const int m   = lane & 15;                  // row within tile (A & C low half)
    const int kb  = (lane >> 4) << 1;           // K sub-offset: 0 (lanes 0-15), 2 (16-31)
    const int nn  = lane & 15;                  // column for B/C

    v8f c = {};
    #pragma unroll 4
    for (int k0 = 0; k0 < D_IN; k0 += 4) {
        const float2 ap = *reinterpret_cast<const float2*>(
            x + (mBase + m) * D_IN + k0 + kb);
        v2f a; a.x = ap.x; a.y = ap.y;
        v2f b;
        b.x = W[(k0 + kb)     * D_OUT + nBase + nn];
        b.y = W[(k0 + kb + 1) * D_OUT + nBase + nn];
        c = __builtin_amdgcn_wmma_f32_16x16x4_f32(false, a, false, b,
                                                  (short)0, c, false, false);
    }
    const int rowOff = (lane >> 4) << 3;
    #pragma unroll
    for (int r = 0; r < 8; ++r) {
        xw[(mBase + r + rowOff) * D_OUT + nBase + nn] = c[r];
    }
}

// ---------------------------------------------------------------------------
// GEMM 2: gi[N,384] = g[N,128] @ w_ih^T + b_ih    (w_ih is [384,128])
// ---------------------------------------------------------------------------
__global__ void __launch_bounds__(256)
gemm_gi_kernel(const float* __restrict__ g, const float* __restrict__ w_ih,
               const float* __restrict__ b_ih, float* __restrict__ gi) {
    const int lane  = threadIdx.x & 31;
    const int wave  = threadIdx.x >> 5;
    const int tile  = blockIdx.x * 8 + wave;
    const int nTiles = G3H / 16;                // 24
    const int mTile = tile / nTiles;
    const int nTile = tile % nTiles;
    const int mBase = mTile * 16;
    const int nBase = nTile * 16;

    const int m   = lane & 15;
    const int kb  = (lane >> 4) << 1;
    const int nn  = lane & 15;
    const int col = nBase + nn;

    v8f c = {};
    #pragma unroll 4
    for (int k0 = 0; k0 < H_DIM; k0 += 4) {
        const float2 ap = *reinterpret_cast<const float2*>(
            g + (mBase + m) * H_DIM + k0 + kb);
        v2f a; a.x = ap.x; a.y = ap.y;
        const float2 bp = *reinterpret_cast<const float2*>(
            w_ih + col * H_DIM + k0 + kb);
        v2f b; b.x = bp.x; b.y = bp.y;
        c = __builtin_amdgcn_wmma_f32_16x16x4_f32(false, a, false, b,
                                                  (short)0, c, false, false);
    }
    const float bias = b_ih[col];
    const int rowOff = (lane >> 4) << 3;
    #pragma unroll
    for (int r = 0; r < 8; ++r) {
        gi[(mBase + r + rowOff) * G3H + col] = c[r] + bias;
    }
}

// ---------------------------------------------------------------------------
// Degree / normalization / aggregation
// ---------------------------------------------------------------------------
__global__ void deg_init_kernel(float* __restrict__ deg) {
    int i = blockIdx.x * blockDim.x + threadIdx.x;
    if (i < N_NODES) deg[i] = 1.0f;             // self-loop
}

__global__ void deg_accum_kernel(const int* __restrict__ ei,
                                 float* __restrict__ deg) {
    int e = blockIdx.x * blockDim.x + threadIdx.x;
    if (e < N_EDGES) atomicAdd(&deg[ei[N_EDGES + e]], 1.0f);
}

__global__ void dis_kernel(const float* __restrict__ deg,
                           float* __restrict__ dis) {
    int i = blockIdx.x * blockDim.x + threadIdx.x;
    if (i < N_NODES) dis[i] = __frsqrt_rn(deg[i]);
}

// g[i][d] = bias[d] + dis[i]^2 * xw[i][d]   (self-loop contribution + bias)
__global__ void agg_init_kernel(const float* __restrict__ xw,
                                const float* __restrict__ dis,
                                const float* __restrict__ bias,
                                float* __restrict__ g) {
    int idx = blockIdx.x * blockDim.x + threadIdx.x;   // N_NODES * 128
    int i = idx >> 7;
    int d = idx & 127;
    float di = dis[i];
    g[idx] = bias[d] + di * di * xw[idx];
}

// scatter: g[tgt][d] += dis[src]*dis[tgt] * xw[src][d]
__global__ void edge_agg_kernel(const int* __restrict__ ei,
                                const float* __restrict__ xw,
                                const float* __restrict__ dis,
                                float* __restrict__ g) {
    long long idx = (long long)blockIdx.x * blockDim.x + threadIdx.x;
    int e = (int)(idx >> 7);
    int d = (int)(idx & 127);
    int s = ei[e];
    int t = ei[N_EDGES + e];
    float v = dis[s] * dis[t] * xw[s * D_OUT + d];
    atomicAdd(&g[t * D_OUT + d], v);
}

// ---------------------------------------------------------------------------
// GRU scan: single workgroup, 384 threads (12 wave32 on one WGP).
// - w_hh row j lives in thread j's VGPRs (128 f32).
// - h lives in LDS (broadcast ds reads).
// - gi rows are double-buffered into LDS with GLOBAL_LOAD_ASYNC_TO_LDS_B32
//   (ASYNCcnt-tracked), so the gate phase never touches global memory.
// Gate order r,z,n (torch GRU).
// ---------------------------------------------------------------------------
__device__ __forceinline__ float sigmoidf_(float x) {
    return 1.0f / (1.0f + __expf(-x));
}

__device__ __forceinline__ void async_load_b32_to_lds(const float* gsrc,
                                                      const float* ldst) {
    // Per ISA (CDNA5 ISA): GLOBAL_LOAD_ASYNC_TO_LDS_B32, GV mode.
    // VDST = VGPR holding LDS byte address; VADDR = 64-bit global address.
    // Generic pointers to LDS map addr[31:0] -> LDS byte address (aperture).
    unsigned lds_addr = (unsigned)(unsigned long long)(const void*)ldst;
    unsigned long long gaddr = (unsigned long long)(const void*)gsrc;
    asm volatile("global_load_async_to_lds_b32 %0, %1, off"
                 :
                 : "v"(lds_addr), "v"(gaddr)
                 : "memory");
}

__device__ __forceinline__ void wait_asynccnt0() {
    asm volatile("s_wait_asynccnt 0x0" ::: "memory");
}

__global__ void __launch_bounds__(G3H)
gru_scan_kernel(const float* __restrict__ gi, const float* __restrict__ w_hh,
                const float* __restrict__ b_hh, const float* __restrict__ h0,
                float* __restrict__ ys, float* __restrict__ h_last) {
    __shared__ float hs[H_DIM];
    __shared__ float gh[G3H];
    __shared__ float gibuf[2][G3H];             // double-buffered gi rows

    const int j = threadIdx.x;

    // Pull my w_hh row into registers (32 x float4 = 128 VGPRs of weights).
    float wrow[H_DIM];
    {
        const float4* wv = reinterpret_cast<const float4*>(w_hh + j * H_DIM);
        #pragma unroll
        for (int q = 0; q < H_DIM / 4; ++q) {
            float4 t = wv[q];
            wrow[4 * q + 0] = t.x;
            wrow[4 * q + 1] = t.y;
            wrow[4 * q + 2] = t.z;
            wrow[4 * q + 3] = t.w;
        }
    }
    const float bj = b_hh[j];
    if (j < H_DIM) hs[j] = h0[j];
    gibuf[0][j] = gi[j];                        // stage row 0 synchronously
    __syncthreads();

    for (int t = 0; t < N_NODES; ++t) {
        // Kick off async stage of the NEXT gi row into the other LDS buffer;
        // it lands while we do this step's matvec + gates.
        if (t + 1 < N_NODES)
            async_load_b32_to_lds(gi + (long long)(t + 1) * G3H + j,
                                  &gibuf[(t + 1) & 1][j]);

        // gh[j] = b_hh[j] + dot(h, w_hh[j]) — broadcast ds loads of hs[k],
        // weights from VGPRs, fully unrolled FMA chain.
        float acc = bj;
        #pragma unroll
        for (int k = 0; k < H_DIM; k += 4) {
            acc += hs[k + 0] * wrow[k + 0];
            acc += hs[k + 1] * wrow[k + 1];
            acc += hs[k + 2] * wrow[k + 2];
            acc += hs[k + 3] * wrow[k + 3];
        }
        gh[j] = acc;
        __syncthreads();

        if (j < H_DIM) {
            const float* git = gibuf[t & 1];
            float r  = sigmoidf_(git[j]             + gh[j]);
            float z  = sigmoidf_(git[H_DIM + j]     + gh[H_DIM + j]);
            float nn = tanhf    (git[2 * H_DIM + j] + r * gh[2 * H_DIM + j]);
            float hn = (1.0f - z) * nn + z * hs[j];
            hs[j] = hn;
            ys[(long long)t * H_DIM + j] = hn;
        }
        // Drain this wave's outstanding async (row t+1), then barrier so every
        // wave sees all landed data + the hs update before the next step.
        wait_asynccnt0();
        __syncthreads();
    }

    if (j < H_DIM) h_last[j] = hs[j];
}

// ---------------------------------------------------------------------------
// Host launcher
// ---------------------------------------------------------------------------
extern "C" void kernel_launch(void* const* d_in, const int* in_sizes, int n_in,
                              void* d_out, int out_size, void* d_ws, size_t ws_size,
                              hipStream_t stream) {
    (void)in_sizes; (void)n_in; (void)out_size; (void)ws_size;

    const float* x     = (const float*)d_in[0];
    const int*   ei    = (const int*)  d_in[1];   // [2, E]: row0 src, row1 tgt
    const float* gcn_w = (const float*)d_in[2];   // [128,128]
    const float* gcn_b = (const float*)d_in[3];   // [128]
    const float* w_ih  = (const float*)d_in[4];   // [384,128]
    const float* w_hh  = (const float*)d_in[5];   // [384,128]
    const float* b_ih  = (const float*)d_in[6];   // [384]
    const float* b_hh  = (const float*)d_in[7];   // [384]
    const float* h0    = (const float*)d_in[8];   // [1,1,128]

    float* out    = (float*)d_out;
    float* ys     = out;                          // [8192,128]
    float* h_last = out + (size_t)N_NODES * H_DIM;

    // Workspace partition (floats): xw | g | gi | dis | deg
    float* xw  = (float*)d_ws;
    float* g   = xw + (size_t)N_NODES * D_OUT;
    float* gi  = g  + (size_t)N_NODES * D_OUT;
    float* dis = gi + (size_t)N_NODES * G3H;
    float* deg = dis + N_NODES;

    // 1) xw = x @ W  (WMMA f32). Tiles: 512 x 8 = 4096 waves, 8 waves/block.
    gemm_xw_kernel<<<(N_NODES / 16) * (D_OUT / 16) / 8, 256, 0, stream>>>(x, gcn_w, xw);

    // 2) degrees (self-loops + edge targets), then dis = rsqrt(deg)
    deg_init_kernel<<<(N_NODES + 255) / 256, 256, 0, stream>>>(deg);
    deg_accum_kernel<<<(N_EDGES + 255) / 256, 256, 0, stream>>>(ei, deg);
    dis_kernel<<<(N_NODES + 255) / 256, 256, 0, stream>>>(deg, dis);

    // 3) g = bias + dis^2 * xw  (self loop), then scatter edges
    agg_init_kernel<<<(N_NODES * D_OUT) / 256, 256, 0, stream>>>(xw, dis, gcn_b, g);
    edge_agg_kernel<<<((long long)N_EDGES * D_OUT) / 256, 256, 0, stream>>>(ei, xw, dis, g);

    // 4) gi = g @ w_ih^T + b_ih  (WMMA f32). Tiles: 512 x 24, 8 waves/block.
    gemm_gi_kernel<<<(N_NODES / 16) * (G3H / 16) / 8, 256, 0, stream>>>(g, w_ih, b_ih, gi);

    // 5) sequential GRU over 8192 steps, single WGP-resident workgroup.
    gru_scan_kernel<<<1, G3H, 0, stream>>>(gi, w_hh, b_hh, h0, ys, h_last);
}